// UserEncoder_10299331576091
// MI455X (gfx1250) — compile-verified
//
#include <hip/hip_runtime.h>
#include <math.h>

typedef float v2f __attribute__((ext_vector_type(2)));
typedef float v4f __attribute__((ext_vector_type(4)));
typedef float v8f __attribute__((ext_vector_type(8)));

#define BB 64
#define NN 32
#define HH 100
#define DD 400
#define AA 200
#define W1_LD (2 * DD) /* 800 */

__device__ __forceinline__ float fast_tanh(float x) {
#if __has_builtin(__builtin_amdgcn_tanhf)
    return __builtin_amdgcn_tanhf(x);   // v_tanh_f32 on gfx1250
#else
    return tanhf(x);
#endif
}

// ---------------------------------------------------------------------------
// Kernel 1: P[M x 200] = X[M x 400] * W1[:, wcol_off : wcol_off+400]^T (+bias)
// One wave per 32x16 output tile: two 16x16 accumulators share each B
// fragment (3 b64 loads : 2 v_wmma_f32_16x16x4_f32 per K-step).
// A frag (16x4 f32): lanes 0-15 hold M=lane, K={0,1}; lanes 16-31 K={2,3}.
// B frag (4x16 f32): lanes 0-15 hold N=lane, K={0,1}; lanes 16-31 K={2,3}.
// D (16x16 f32):    VGPR r -> row r (lanes 0-15) / row r+8 (lanes 16-31).
// ---------------------------------------------------------------------------
__global__ __launch_bounds__(128)
void proj_wmma_kernel(const float* __restrict__ X,
                      const float* __restrict__ W1,
                      const float* __restrict__ bias,
                      float* __restrict__ P,
                      int tilesM, int tilesN, int wcol_off)
{
    int tile  = blockIdx.x * blockDim.y + threadIdx.y;
    int total = tilesM * tilesN;
    if (tile >= total) return;
    int tm = tile / tilesN;   // 32-row tile index
    int tn = tile - tm * tilesN;

    int lane = threadIdx.x;
    int half = lane >> 4;     // which K-pair this lane holds
    int lr   = lane & 15;

    int n  = tn * 16 + lr;                  // B column (ragged at N=200)
    int nc = (n < AA) ? n : (AA - 1);       // clamp for OOB-safe loads; stores guarded

    const float* a0   = X  + (size_t)(tm * 32 + lr) * DD + 2 * half;
    const float* a1   = a0 + (size_t)16 * DD;
    const float* bptr = W1 + (size_t)nc * W1_LD + wcol_off + 2 * half;

    v8f c0 = {0.f, 0.f, 0.f, 0.f, 0.f, 0.f, 0.f, 0.f};
    v8f c1 = c0;
#pragma unroll 4
    for (int k = 0; k < DD; k += 4) {
        v2f b  = *(const v2f*)(bptr + k);   // shared B fragment
        v2f x0 = *(const v2f*)(a0 + k);
        v2f x1 = *(const v2f*)(a1 + k);
        c0 = __builtin_amdgcn_wmma_f32_16x16x4_f32(
                 false, x0, false, b, (short)0, c0, false, false);
        c1 = __builtin_amdgcn_wmma_f32_16x16x4_f32(
                 false, x1, false, b, (short)0, c1, false, false);
    }

    if (n < AA) {
        float bv = bias ? bias[n] : 0.0f;
#pragma unroll
        for (int r = 0; r < 8; ++r) {
            int row0 = tm * 32 + r + 8 * half;
            P[(size_t)row0 * AA + n]        = c0[r] + bv;
            P[(size_t)(row0 + 16) * AA + n] = c1[r] + bv;
        }
    }
}

// ---------------------------------------------------------------------------
// Kernel 2: per (b,n) block: logits over H via tanh-dot with W2, mask,
// softmax over H, then out[b,n,:] = attn . log_vec[b,:,:].
// ---------------------------------------------------------------------------
__global__ __launch_bounds__(128)
void attn_softmax_out_kernel(const float* __restrict__ pn,
                             const float* __restrict__ pl,
                             const float* __restrict__ W2,
                             const float* __restrict__ b2,
                             const int*   __restrict__ log_mask,
                             const float* __restrict__ log_vec,
                             float* __restrict__ out)
{
    __shared__ float pn_s[AA];
    __shared__ float w2_s[AA];
    __shared__ float attn_s[HH];
    __shared__ float red[128];

    int t  = threadIdx.x;
    int bn = blockIdx.x;          // 0 .. B*N-1
    int b  = bn / NN;

    const float* pn_row = pn + (size_t)bn * AA;
    for (int i = t; i < AA; i += 128) {
        pn_s[i] = pn_row[i];
        w2_s[i] = W2[i];
    }
    __syncthreads();

    float logit = -1.0e9f;        // matches reference mask fill
    if (t < HH) {
        if (log_mask[b * HH + t] != 0) {
            const float* pl_row = pl + ((size_t)b * HH + t) * AA;
            float acc = 0.0f;
#pragma unroll 2
            for (int a = 0; a < AA; a += 4) {
                v4f p = *(const v4f*)(pl_row + a);   // 16B-aligned b128 load
                acc += w2_s[a + 0] * fast_tanh(pn_s[a + 0] + p.x);
                acc += w2_s[a + 1] * fast_tanh(pn_s[a + 1] + p.y);
                acc += w2_s[a + 2] * fast_tanh(pn_s[a + 2] + p.z);
                acc += w2_s[a + 3] * fast_tanh(pn_s[a + 3] + p.w);
            }
            logit = acc + b2[0];
        }
    }

    // max-reduce over H
    red[t] = (t < HH) ? logit : -INFINITY;
    __syncthreads();
    for (int s = 64; s > 0; s >>= 1) {
        if (t < s) red[t] = fmaxf(red[t], red[t + s]);
        __syncthreads();
    }
    float mx = red[0];
    __syncthreads();

    // exp + sum-reduce
    float e = (t < HH) ? __expf(logit - mx) : 0.0f;
    red[t] = e;
    __syncthreads();
    for (int s = 64; s > 0; s >>= 1) {
        if (t < s) red[t] += red[t + s];
        __syncthreads();
    }
    float denom = red[0];
    if (t < HH) attn_s[t] = e / denom;
    __syncthreads();

    // out[bn, d] = sum_h attn[h] * log_vec[b, h, d]  (coalesced over d)
    const float* lv   = log_vec + (size_t)b * HH * DD;
    float*       orow = out + (size_t)bn * DD;
    for (int d = t; d < DD; d += 128) {
        float acc = 0.0f;
#pragma unroll 4
        for (int h = 0; h < HH; ++h)
            acc += attn_s[h] * lv[(size_t)h * DD + d];
        orow[d] = acc;
    }
}

// ---------------------------------------------------------------------------
extern "C" void kernel_launch(void* const* d_in, const int* in_sizes, int n_in,
                              void* d_out, int out_size, void* d_ws, size_t ws_size,
                              hipStream_t stream)
{
    const float* log_vec  = (const float*)d_in[0];
    const int*   log_mask = (const int*)  d_in[1];
    const float* news_vec = (const float*)d_in[2];
    const float* W1       = (const float*)d_in[3];
    const float* b1       = (const float*)d_in[4];
    const float* W2       = (const float*)d_in[5];
    const float* b2       = (const float*)d_in[6];
    float*       out      = (float*)d_out;

    // workspace: pn (B*N x A) then pl (B*H x A)  => 6.76 MB total
    float* pn = (float*)d_ws;
    float* pl = pn + (size_t)BB * NN * AA;

    dim3 blk(32, 4); // 4 waves per block, one 32x16 tile per wave

    { // pn = news_vec . Wn^T          (M = 2048 -> 64 row-tiles of 32)
        int tilesM = (BB * NN) / 32;
        int tilesN = (AA + 15) / 16;
        int total  = tilesM * tilesN;
        proj_wmma_kernel<<<(total + 3) / 4, blk, 0, stream>>>(
            news_vec, W1, nullptr, pn, tilesM, tilesN, /*wcol_off=*/0);
    }
    { // pl = log_vec . Wl^T + b1      (M = 6400 -> 200 row-tiles of 32)
        int tilesM = (BB * HH) / 32;
        int tilesN = (AA + 15) / 16;
        int total  = tilesM * tilesN;
        proj_wmma_kernel<<<(total + 3) / 4, blk, 0, stream>>>(
            log_vec, W1, b1, pl, tilesM, tilesN, /*wcol_off=*/DD);
    }
    // fused tanh-attention + softmax + weighted sum
    attn_softmax_out_kernel<<<BB * NN, 128, 0, stream>>>(
        pn, pl, W2, b2, log_mask, log_vec, out);
}